// Net_68118181314921
// MI455X (gfx1250) — compile-verified
//
#include <hip/hip_runtime.h>
#include <hip/hip_bf16.h>

#define N_NODES  100000
#define N_EDGES  3200000
#define N_GRAPHS 256
#define F_IN     128

typedef __attribute__((ext_vector_type(2))) float v2f;
typedef __attribute__((ext_vector_type(8))) float v8f;

__device__ __forceinline__ float elu_f(float v) {
    return v > 0.f ? v : (__expf(v) - 1.f);
}

// ---------------------------------------------------------------------------
// Zero a float buffer (workspace must be re-zeroed every call; harness does
// not re-initialize d_ws between replays).
// ---------------------------------------------------------------------------
__global__ void zero_f32(float* __restrict__ p, int n) {
    int i = blockIdx.x * blockDim.x + threadIdx.x;
    if (i < n) p[i] = 0.f;
}

// ---------------------------------------------------------------------------
// h = x @ W1   (100000x128 @ 128x16), fp32 WMMA 16x16x4.
// One wave (32 lanes) per 16-row tile; 6250 tiles exactly.
// A frag (16x4 f32): lanes 0-15 -> M=lane, K=k0..k0+1; lanes 16-31 -> M=lane-16,
// K=k0+2..k0+3.  B frag (4x16): VGPR0 -> K={k0,k0+2}, VGPR1 -> K={k0+1,k0+3}.
// C/D (16x16 f32): VGPR r -> M = r + (lane>=16 ? 8 : 0), N = lane&15.
// ---------------------------------------------------------------------------
__global__ void gemm_xw1(const float* __restrict__ x,
                         const float* __restrict__ W1,
                         float* __restrict__ h) {
    const int tile = blockIdx.x;          // 0..6249
    const int lane = threadIdx.x;         // 0..31
    const int half = lane >> 4;           // 0 | 1
    const int l    = lane & 15;
    const int row  = tile * 16 + l;

    v8f c = {};
    const float* xrow = x + (size_t)row * F_IN + half * 2;
#pragma unroll 4
    for (int k0 = 0; k0 < F_IN; k0 += 4) {
        v2f a = *(const v2f*)(xrow + k0);                 // K = k0+2h, k0+2h+1
        v2f b;
        b[0] = W1[(k0 + 2 * half) * 16 + l];
        b[1] = W1[(k0 + 2 * half + 1) * 16 + l];
        c = __builtin_amdgcn_wmma_f32_16x16x4_f32(false, a, false, b,
                                                  (short)0, c, false, false);
    }
    float* out = h + (size_t)(tile * 16 + half * 8) * 16 + l;
#pragma unroll
    for (int r = 0; r < 8; ++r) out[r * 16] = c[r];
}

// ---------------------------------------------------------------------------
// Edge scatter:  agg[row[e]] += h[col[e]] * w[e]
// 4 lanes per edge, each lane owns a float4 channel group -> coalesced 64B
// gathers and clustered atomic targets. agg (6.4MB) is L2-resident.
// ---------------------------------------------------------------------------
__global__ void edge_scatter(const float* __restrict__ h,
                             const float* __restrict__ w,
                             const int* __restrict__ row,
                             const int* __restrict__ col,
                             float* __restrict__ agg) {
    int tid = blockIdx.x * blockDim.x + threadIdx.x;
    int e = tid >> 2;
    if (e >= N_EDGES) return;
    int cg = (tid & 3) * 4;

    float we  = w[e];
    int   src = col[e];
    int   dst = row[e];

    float4 v = *(const float4*)(h + (size_t)src * 16 + cg);
    float* ap = agg + (size_t)dst * 16 + cg;
    atomicAdd(ap + 0, v.x * we);
    atomicAdd(ap + 1, v.y * we);
    atomicAdd(ap + 2, v.z * we);
    atomicAdd(ap + 3, v.w * we);
}

// ---------------------------------------------------------------------------
// h2 = elu(agg + b1) @ W2   (100000x16 @ 16x16), fp32 WMMA, K=16 -> 4 steps.
// bias+ELU fused into the A-fragment load (each element consumed once).
// ---------------------------------------------------------------------------
__global__ void gemm_layer2(const float* __restrict__ agg,
                            const float* __restrict__ b1,
                            const float* __restrict__ W2,
                            float* __restrict__ h2) {
    const int tile = blockIdx.x;
    const int lane = threadIdx.x;
    const int half = lane >> 4;
    const int l    = lane & 15;
    const int row  = tile * 16 + l;

    v8f c = {};
#pragma unroll
    for (int kk = 0; kk < 4; ++kk) {
        int k0 = kk * 4 + half * 2;
        float u0 = elu_f(agg[(size_t)row * 16 + k0]     + b1[k0]);
        float u1 = elu_f(agg[(size_t)row * 16 + k0 + 1] + b1[k0 + 1]);
        v2f a; a[0] = u0; a[1] = u1;
        v2f b;
        b[0] = W2[k0 * 16 + l];
        b[1] = W2[(k0 + 1) * 16 + l];
        c = __builtin_amdgcn_wmma_f32_16x16x4_f32(false, a, false, b,
                                                  (short)0, c, false, false);
    }
    float* out = h2 + (size_t)(tile * 16 + half * 8) * 16 + l;
#pragma unroll
    for (int r = 0; r < 8; ++r) out[r * 16] = c[r];
}

// ---------------------------------------------------------------------------
// pooled[seg[n]] += elu(agg[n] + b2)   — 1.6M atomics into a 16KB buffer.
// ---------------------------------------------------------------------------
__global__ void pool_nodes(const float* __restrict__ agg,
                           const float* __restrict__ b2,
                           const int* __restrict__ seg,
                           float* __restrict__ pooled) {
    int tid = blockIdx.x * blockDim.x + threadIdx.x;
    int n = tid >> 4;
    if (n >= N_NODES) return;
    int cch = tid & 15;
    float u = elu_f(agg[tid] + b2[cch]);
    atomicAdd(&pooled[seg[n] * 16 + cch], u);
}

// ---------------------------------------------------------------------------
// out = sigmoid(relu(pooled @ Wd2 + bd2) @ Wd3 + bd3)    (256 graphs)
// ---------------------------------------------------------------------------
__global__ void head_kernel(const float* __restrict__ pooled,
                            const float* __restrict__ Wd2,
                            const float* __restrict__ bd2,
                            const float* __restrict__ Wd3,
                            const float* __restrict__ bd3,
                            float* __restrict__ out) {
    int g = blockIdx.x * blockDim.x + threadIdx.x;
    if (g >= N_GRAPHS) return;
    float t[16];
#pragma unroll
    for (int j = 0; j < 16; ++j) {
        float s = bd2[j];
#pragma unroll
        for (int k = 0; k < 16; ++k) s += pooled[g * 16 + k] * Wd2[k * 16 + j];
        t[j] = s > 0.f ? s : 0.f;
    }
#pragma unroll
    for (int o = 0; o < 2; ++o) {
        float s = bd3[o];
#pragma unroll
        for (int j = 0; j < 16; ++j) s += t[j] * Wd3[j * 2 + o];
        out[g * 2 + o] = 1.f / (1.f + __expf(-s));
    }
}

// ---------------------------------------------------------------------------
extern "C" void kernel_launch(void* const* d_in, const int* in_sizes, int n_in,
                              void* d_out, int out_size, void* d_ws, size_t ws_size,
                              hipStream_t stream) {
    const float* x   = (const float*)d_in[0];
    const float* ew  = (const float*)d_in[1];
    const float* W1  = (const float*)d_in[2];
    const float* b1  = (const float*)d_in[3];
    const float* W2  = (const float*)d_in[4];
    const float* b2  = (const float*)d_in[5];
    const float* Wd2 = (const float*)d_in[6];
    const float* bd2 = (const float*)d_in[7];
    const float* Wd3 = (const float*)d_in[8];
    const float* bd3 = (const float*)d_in[9];
    const int*   row = (const int*)d_in[10];
    const int*   col = (const int*)d_in[11];
    const int*   seg = (const int*)d_in[12];
    float* out = (float*)d_out;

    // Workspace layout: h | agg | pooled (contiguous)
    float* h      = (float*)d_ws;                 // N_NODES*16
    float* agg    = h + (size_t)N_NODES * 16;     // N_NODES*16
    float* pooled = agg + (size_t)N_NODES * 16;   // N_GRAPHS*16

    const int NT = N_NODES / 16;                  // 6250 exact 16-row tiles
    const int NAGG = N_NODES * 16;

    // Layer 1
    zero_f32<<<(NAGG + 255) / 256, 256, 0, stream>>>(agg, NAGG);
    gemm_xw1<<<NT, 32, 0, stream>>>(x, W1, h);
    edge_scatter<<<(N_EDGES * 4 + 255) / 256, 256, 0, stream>>>(h, ew, row, col, agg);

    // Layer 2 (h buffer reused for pre-aggregation features)
    gemm_layer2<<<NT, 32, 0, stream>>>(agg, b1, W2, h);
    zero_f32<<<(NAGG + N_GRAPHS * 16 + 255) / 256, 256, 0, stream>>>(agg, NAGG + N_GRAPHS * 16);
    edge_scatter<<<(N_EDGES * 4 + 255) / 256, 256, 0, stream>>>(h, ew, row, col, agg);

    // Pool + head
    pool_nodes<<<(N_NODES * 16 + 255) / 256, 256, 0, stream>>>(agg, b2, seg, pooled);
    head_kernel<<<1, 256, 0, stream>>>(pooled, Wd2, bd2, Wd3, bd3, out);
}